// TreeCRF_54975581389324
// MI455X (gfx1250) — compile-verified
//
#include <hip/hip_runtime.h>
#include <hip/hip_bf16.h>
#include <stdint.h>

// CYK inside algorithm (log semiring), B=128, L=256, on MI455X (gfx1250).
// - One workgroup per batch element; the entire 256x256 f32 DP table (256KB)
//   lives in CDNA5's 320KB-per-workgroup LDS.
// - Per-step diagonal of s_span is DMA'd into LDS by the Tensor Data Mover
//   (strided 2D descriptor), double-buffered and overlapped with compute
//   (TENSORcnt <= 1 pipelining).
// - 512 threads (16 wave32s): the k-reduction of each cell is split across two
//   thread halves, partial (max, sum) pairs merged via a logsumexp combine.

#define NEGV -1000000000.0f
constexpr int LSEQ = 256;
constexpr int NTHR = 2 * LSEQ;   // 512 threads = 16 wave32s

typedef __attribute__((ext_vector_type(4))) unsigned int v4u;
typedef __attribute__((ext_vector_type(8))) unsigned int v8u;

// TDM: load `count` f32 elements from gptr with element stride `stride_elems`
// (a matrix diagonal) into contiguous LDS at lds_byte_off.
// D# per CDNA5 ISA ch.8: group0 (4 dw) + group1 (8 dw); 2D tensor -> groups
// 2/3 NULL, 2-operand asm form.
__device__ __forceinline__ void tdm_load_diag(const float* gptr,
                                              uint32_t lds_byte_off,
                                              uint32_t count,
                                              uint32_t stride_elems) {
  uint64_t ga = (uint64_t)(uintptr_t)gptr;

  v4u g0;
  g0[0] = 1u;                                  // count=1 (valid user descriptor)
  g0[1] = lds_byte_off;                        // lds_addr (bytes)
  g0[2] = (uint32_t)(ga & 0xFFFFFFFFu);        // global_addr[31:0]
  g0[3] = (uint32_t)((ga >> 32) & 0x01FFFFFFu) // global_addr[56:32]
          | (2u << 30);                        // type = 2 ("image")

  v8u g1;
  g1[0] = (2u << 16);                               // data_size=2 (4B); wg_mask=0
  g1[1] = ((uint32_t)LSEQ & 0xFFFFu) << 16;         // tensor_dim0[15:0]
  g1[2] = (((uint32_t)LSEQ >> 16) & 0xFFFFu)        // tensor_dim0[31:16]
          | ((count & 0xFFFFu) << 16);              // tensor_dim1[15:0] = count
  g1[3] = ((count >> 16) & 0xFFFFu)                 // tensor_dim1[31:16]
          | (1u << 16);                             // tile_dim0 = 1
  g1[4] = (count & 0xFFFFu);                        // tile_dim1 = count; tile_dim2=0
  g1[5] = stride_elems;                             // tensor_dim0_stride[31:0]
  g1[6] = 0u;
  g1[7] = 0u;

  asm volatile("tensor_load_to_lds %0, %1" :: "s"(g0), "s"(g1) : "memory");
}

__global__ __launch_bounds__(NTHR) void cyk_inside_kernel(
    const float* __restrict__ s_span,   // [B, L, L]
    const int*   __restrict__ lens,     // [B]
    float*       __restrict__ out) {    // [B]
  __shared__ float sd[LSEQ * LSEQ];     // 256 KB DP table (diag layout)
  __shared__ float spanw[2 * LSEQ];     // double-buffered TDM landing zone
  __shared__ float pm[2 * LSEQ];        // per-half partial max
  __shared__ float ps[2 * LSEQ];        // per-half partial sum

  const int b    = blockIdx.x;
  const int tid  = threadIdx.x;
  const int i    = tid & (LSEQ - 1);    // column owned by this thread
  const int half = tid >> 8;            // which k-half this thread reduces
  const float* __restrict__ sb = s_span + (size_t)b * LSEQ * LSEQ;

  // Row w=1 init: sd[1][i] = s_span[b, i, i+1] for i < L-1, else NEG.
  if (half == 0) {
    float v = (i < LSEQ - 1) ? sb[(size_t)i * LSEQ + (i + 1)] : NEGV;
    sd[1 * LSEQ + i] = v;
  }

  // LDS byte offset of the span buffers (low 32 bits of the generic address of
  // an LDS object = offset within the workgroup allocation).
  const uint32_t span_lds = (uint32_t)(uintptr_t)(&spanw[0]);

  // Prologue: DMA diagonal for w=2 into buffer 0 (2&1 == 0).
  if (tid == 0) {
    tdm_load_diag(sb + 2, span_lds, (uint32_t)(LSEQ - 2), (uint32_t)(LSEQ + 1));
  }

  for (int w = 2; w < LSEQ; ++w) {
    __syncthreads();   // [A] step w-1 fully done: row w-1 written, buffers free

    if (tid == 0) {
      if (w + 1 < LSEQ) {
        // Prefetch diagonal for w+1 into the other buffer; only require the
        // *older* transfer (for w) to have landed -> DMA overlaps compute.
        tdm_load_diag(sb + (w + 1),
                      span_lds + (uint32_t)(((w + 1) & 1) * LSEQ * 4),
                      (uint32_t)(LSEQ - (w + 1)), (uint32_t)(LSEQ + 1));
        __builtin_amdgcn_s_wait_tensorcnt(1);
      } else {
        __builtin_amdgcn_s_wait_tensorcnt(0);
      }
    }
    __syncthreads();   // [B] spanw[w&1] visible to all waves

    // Phase 1: each half reduces its k-range with a two-pass stable logsumexp.
    if (i < LSEQ - w) {
      const int klo = (half == 0) ? 1 : (w / 2 + 1);
      const int khi = (half == 0) ? (w / 2) : (w - 1);
      float m = NEGV;
      #pragma unroll 4
      for (int k = klo; k <= khi; ++k) {
        float v = sd[k * LSEQ + i] + sd[(w - k) * LSEQ + (i + k)];
        m = fmaxf(m, v);
      }
      float s = 0.0f;
      #pragma unroll 4
      for (int k = klo; k <= khi; ++k) {
        float v = sd[k * LSEQ + i] + sd[(w - k) * LSEQ + (i + k)];
        s += __expf(v - m);
      }
      pm[half * LSEQ + i] = m;
      ps[half * LSEQ + i] = s;
    }
    __syncthreads();   // [C] partials visible

    // Phase 2: merge the two (m, s) partials, add span score, write row w.
    if (half == 0) {
      float res = NEGV;
      if (i < LSEQ - w) {
        float m0 = pm[i],        m1 = pm[LSEQ + i];
        float s0 = ps[i],        s1 = ps[LSEQ + i];
        float m  = fmaxf(m0, m1);
        // Empty half (w==2) has s1==0, m1==NEG: exp underflows to 0 -> no NaN.
        float s  = s0 * __expf(m0 - m) + s1 * __expf(m1 - m);
        res = m + __logf(s) + spanw[(w & 1) * LSEQ + i];
      }
      sd[w * LSEQ + i] = res;   // row w first read at step > w (barrier A)
    }
  }

  __syncthreads();
  if (tid == 0) {
    const int len = lens[b];
    out[b] = sd[len * LSEQ + 0];
  }
}

extern "C" void kernel_launch(void* const* d_in, const int* in_sizes, int n_in,
                              void* d_out, int out_size, void* d_ws, size_t ws_size,
                              hipStream_t stream) {
  const float* s_span = (const float*)d_in[0];
  const int*   lens   = (const int*)d_in[1];
  float*       out    = (float*)d_out;

  const int B = in_sizes[1];   // lens has B elements; in_sizes[0] == B*L*L

  cyk_inside_kernel<<<B, NTHR, 0, stream>>>(s_span, lens, out);
}